// TtMambaSSM_44238163149467
// MI455X (gfx1250) — compile-verified
//
#include <hip/hip_runtime.h>
#include <hip/hip_bf16.h>

#define B_SZ    1024
#define D_INNER 2560
#define D_STATE 16
#define DT_RANK 160

typedef __attribute__((ext_vector_type(2))) float v2f;
typedef __attribute__((ext_vector_type(4))) float f4;
typedef __attribute__((ext_vector_type(8))) float v8f;

__device__ __forceinline__ v2f ld2(const float* p) {
  v2f r; r.x = p[0]; r.y = p[1]; return r;
}

// Accumulate one 16x16 f32 tile: acc += A[16xK] * Bw^T, where A is row-major
// (rows starting at Arow, leading dim lda) and Bw is [N,K] row-major (rows at
// Brow, leading dim ldb) -- i.e. we multiply by its transpose, matching the
// 'bd,nd->bn' einsums.
// Fragment layout per CDNA5 ISA 7.12.2 (wave32):
//   A 16x4 f32: lane%16 = M, lane/16 selects K-pair {0,1} vs {2,3}; 2 VGPRs.
//   B 4x16 f32: lane%16 = N, lane/16 selects K-pair; 2 VGPRs (row-striped).
__device__ __forceinline__ void gemm_tile_acc(const float* __restrict__ Arow,
                                              const float* __restrict__ Brow,
                                              int lda, int ldb, int K, v8f& acc) {
  const int lane = threadIdx.x & 31;
  const int l16  = lane & 15;
  const int half = lane >> 4;
  const float* ap = Arow + (size_t)l16 * lda + 2 * half;
  const float* bp = Brow + (size_t)l16 * ldb + 2 * half;
#pragma unroll 4
  for (int k = 0; k < K; k += 4) {
    v2f a = ld2(ap + k);
    v2f b = ld2(bp + k);
    acc = __builtin_amdgcn_wmma_f32_16x16x4_f32(false, a, false, b, (short)0,
                                                acc, false, false);
  }
}

// ---------------------------------------------------------------------------
// Kernel 1: A = -exp(A_log)   [D_INNER, D_STATE]
// ---------------------------------------------------------------------------
__global__ void k_negexp(const float* __restrict__ A_log,
                         float* __restrict__ A, int n) {
  int i = blockIdx.x * blockDim.x + threadIdx.x;
  if (i < n) A[i] = -__expf(A_log[i]);
}

// ---------------------------------------------------------------------------
// Kernel 2: dt_low = x @ W_dt_low^T   [B, DT_RANK], K = D_INNER
// 4 waves/block, one 16x16 tile per wave.
// ---------------------------------------------------------------------------
__global__ void k_gemm_dtlow(const float* __restrict__ x,
                             const float* __restrict__ Wlow,
                             float* __restrict__ dtlow) {
  const int wave = threadIdx.x >> 5;
  const int m0 = (blockIdx.y * 4 + wave) * 16;
  const int n0 = blockIdx.x * 16;
  v8f acc = {};
  gemm_tile_acc(x + (size_t)m0 * D_INNER, Wlow + (size_t)n0 * D_INNER,
                D_INNER, D_INNER, D_INNER, acc);
  const int lane = threadIdx.x & 31;
  const int l16 = lane & 15, half = lane >> 4;
#pragma unroll
  for (int v = 0; v < 8; v++)  // C/D: VGPR v, lanes0-15 -> M=v, lanes16-31 -> M=v+8
    dtlow[(size_t)(m0 + v + 8 * half) * DT_RANK + n0 + l16] = acc[v];
}

// ---------------------------------------------------------------------------
// Kernel 3: dt = softplus(dt_low @ W_dt^T + b_dt)   [B, D_INNER], K = DT_RANK
// ---------------------------------------------------------------------------
__global__ void k_gemm_dt(const float* __restrict__ dtlow,
                          const float* __restrict__ Wdt,
                          const float* __restrict__ bdt,
                          float* __restrict__ dt) {
  const int wave = threadIdx.x >> 5;
  const int m0 = (blockIdx.y * 4 + wave) * 16;
  const int n0 = blockIdx.x * 16;
  v8f acc = {};
  gemm_tile_acc(dtlow + (size_t)m0 * DT_RANK, Wdt + (size_t)n0 * DT_RANK,
                DT_RANK, DT_RANK, DT_RANK, acc);
  const int lane = threadIdx.x & 31;
  const int l16 = lane & 15, half = lane >> 4;
  const float bias = bdt[n0 + l16];   // N is fixed per lane across all 8 regs
#pragma unroll
  for (int v = 0; v < 8; v++) {
    float z = acc[v] + bias;
    float sp = (z > 20.0f) ? z : __logf(1.0f + __expf(z));  // softplus
    dt[(size_t)(m0 + v + 8 * half) * D_INNER + n0 + l16] = sp;
  }
}

// ---------------------------------------------------------------------------
// Kernel 4: Bt = x @ W_B^T, Ct = x @ W_C^T   [B, 16], K = D_INNER
// One N-tile; two accumulators share the A fragment.
// ---------------------------------------------------------------------------
__global__ void k_gemm_bc(const float* __restrict__ x,
                          const float* __restrict__ WB,
                          const float* __restrict__ WC,
                          float* __restrict__ Bt,
                          float* __restrict__ Ct) {
  const int wave = threadIdx.x >> 5;
  const int m0 = (blockIdx.x * 4 + wave) * 16;
  const int lane = threadIdx.x & 31;
  const int l16 = lane & 15, half = lane >> 4;
  v8f accB = {}, accC = {};
  const float* ap  = x  + ((size_t)m0 + l16) * D_INNER + 2 * half;
  const float* bpB = WB + (size_t)l16 * D_INNER + 2 * half;
  const float* bpC = WC + (size_t)l16 * D_INNER + 2 * half;
#pragma unroll 4
  for (int k = 0; k < D_INNER; k += 4) {
    v2f a  = ld2(ap + k);
    v2f bb = ld2(bpB + k);
    v2f bc = ld2(bpC + k);
    accB = __builtin_amdgcn_wmma_f32_16x16x4_f32(false, a, false, bb, (short)0,
                                                 accB, false, false);
    accC = __builtin_amdgcn_wmma_f32_16x16x4_f32(false, a, false, bc, (short)0,
                                                 accC, false, false);
  }
#pragma unroll
  for (int v = 0; v < 8; v++) {
    Bt[(size_t)(m0 + v + 8 * half) * D_STATE + l16] = accB[v];
    Ct[(size_t)(m0 + v + 8 * half) * D_STATE + l16] = accC[v];
  }
}

// ---------------------------------------------------------------------------
// Kernel 5 (bandwidth kernel): y[b,d] = sum_n exp(dt*A[d,n]) * h[b,d,n] * Ct[b,n]
//                                     + dt*x*(Bt[b].Ct[b]) + D[d]*x
// One block per b; each lane streams 16 contiguous h floats per d with
// non-temporal b128 loads (h is read exactly once -> keep L2 for A reuse).
// blockDim must be 256.
// ---------------------------------------------------------------------------
__global__ void k_ssm_update(const float* __restrict__ x,
                             const float* __restrict__ h,
                             const float* __restrict__ dt,
                             const float* __restrict__ Bt,
                             const float* __restrict__ Ct,
                             const float* __restrict__ Aneg,
                             const float* __restrict__ Dv,
                             float* __restrict__ out) {
  const int b = blockIdx.x;
  __shared__ float sCt[D_STATE];
  __shared__ float sBC;
  const int tid = threadIdx.x;
  if (tid < D_STATE) sCt[tid] = Ct[(size_t)b * D_STATE + tid];
  __syncthreads();
  if (tid == 0) {
    float s = 0.f;
#pragma unroll
    for (int n = 0; n < D_STATE; n++)
      s += Bt[(size_t)b * D_STATE + n] * sCt[n];
    sBC = s;   // rank-1 term: sum_n Bt*Ct collapses to a per-b scalar
  }
  __syncthreads();

#pragma unroll
  for (int it = 0; it < D_INNER / 256; ++it) {
    const int d = it * 256 + tid;
    const size_t bd = (size_t)b * D_INNER + d;
    const f4* hp = (const f4*)(h + bd * D_STATE);
    const f4* ap = (const f4*)(Aneg + (size_t)d * D_STATE);
    const float dtv = dt[bd];
    const float xv  = x[bd];
    float acc = 0.f;
#pragma unroll
    for (int q = 0; q < 4; q++) {
      f4 hv = __builtin_nontemporal_load(hp + q);  // stream-once: NT hint
      f4 av = ap[q];                               // A is hot in L2
      acc += __expf(dtv * av.x) * hv.x * sCt[4 * q + 0];
      acc += __expf(dtv * av.y) * hv.y * sCt[4 * q + 1];
      acc += __expf(dtv * av.z) * hv.z * sCt[4 * q + 2];
      acc += __expf(dtv * av.w) * hv.w * sCt[4 * q + 3];
    }
    out[bd] = acc + dtv * xv * sBC + Dv[d] * xv;
  }
}

// ---------------------------------------------------------------------------
extern "C" void kernel_launch(void* const* d_in, const int* in_sizes, int n_in,
                              void* d_out, int out_size, void* d_ws, size_t ws_size,
                              hipStream_t stream) {
  (void)in_sizes; (void)n_in; (void)out_size; (void)ws_size;
  const float* x    = (const float*)d_in[0];
  const float* h    = (const float*)d_in[1];
  const float* Wlow = (const float*)d_in[2];
  const float* Wdt  = (const float*)d_in[3];
  const float* bdt  = (const float*)d_in[4];
  const float* WB   = (const float*)d_in[5];
  const float* WC   = (const float*)d_in[6];
  const float* Alog = (const float*)d_in[7];
  const float* Dv   = (const float*)d_in[8];
  float* out = (float*)d_out;

  // Workspace layout (floats): 11.4 MB total
  float* w_dtlow = (float*)d_ws;                          // [B, DT_RANK]
  float* w_dt    = w_dtlow + (size_t)B_SZ * DT_RANK;      // [B, D_INNER]
  float* w_Bt    = w_dt    + (size_t)B_SZ * D_INNER;      // [B, D_STATE]
  float* w_Ct    = w_Bt    + (size_t)B_SZ * D_STATE;      // [B, D_STATE]
  float* w_A     = w_Ct    + (size_t)B_SZ * D_STATE;      // [D_INNER, D_STATE]

  k_negexp<<<dim3((D_INNER * D_STATE + 255) / 256), 256, 0, stream>>>(
      Alog, w_A, D_INNER * D_STATE);
  k_gemm_dtlow<<<dim3(DT_RANK / 16, B_SZ / 64), 128, 0, stream>>>(x, Wlow, w_dtlow);
  k_gemm_dt<<<dim3(D_INNER / 16, B_SZ / 64), 128, 0, stream>>>(w_dtlow, Wdt, bdt, w_dt);
  k_gemm_bc<<<dim3(B_SZ / 64), 128, 0, stream>>>(x, WB, WC, w_Bt, w_Ct);
  k_ssm_update<<<dim3(B_SZ), 256, 0, stream>>>(x, h, w_dt, w_Bt, w_Ct, w_A, Dv, out);
}